// SigLossHom_68977174774177
// MI455X (gfx1250) — compile-verified
//
#include <hip/hip_runtime.h>
#include <math.h>

// fp32 WMMA operand types for V_WMMA_F32_16X16X4_F32:
//   A: 16x4 f32  -> 2 VGPRs/lane, B: 4x16 f32 -> 2 VGPRs/lane, C/D: 16x16 f32 -> 8 VGPRs/lane
typedef __attribute__((ext_vector_type(2))) float v2f;
typedef __attribute__((ext_vector_type(8))) float v8f;

#define A_BATCH 64
#define LSEQ    256
#define DDIM    8
#define ZLEN    512   // 2*LSEQ path points
#define LL      511   // number of increments dz
#define NT      512   // threads per workgroup (16 waves of 32)
#define NW      16

// One workgroup per batch element. Entire Goursat solve lives in LDS.
__global__ __launch_bounds__(NT) void SigLoss_pde_kernel(const float* __restrict__ X,
                                                         const float* __restrict__ Y,
                                                         float* __restrict__ partial) {
    __shared__ float dzs[ZLEN][DDIM];   // 16 KB : path increments (row 511 zeroed)
    __shared__ float incBlk[16][NT];    // 32 KB : 16-row block of the Gram matrix
    __shared__ float krow[NT];          //  2 KB : current PDE row K[i][0..511]
    __shared__ float waveSums[NW];

    const int a = blockIdx.x;
    const int t = threadIdx.x;
    const unsigned lane = t & 31u;      // wave32
    const unsigned wave = t >> 5;       // 0..15

    // ---------------- Phase 0: build dz in LDS ----------------
    // Z[l] = X[a,l] for l<256 ; Y[a, 511-l] for l>=256 (flip of Y)
    if (t < LL) {
        const float* Xa = X + (size_t)a * LSEQ * DDIM;
        const float* Ya = Y + (size_t)a * LSEQ * DDIM;
        const int l0 = t, l1 = t + 1;
#pragma unroll
        for (int d = 0; d < DDIM; ++d) {
            float z0 = (l0 < LSEQ) ? Xa[l0 * DDIM + d] : Ya[(2 * LSEQ - 1 - l0) * DDIM + d];
            float z1 = (l1 < LSEQ) ? Xa[l1 * DDIM + d] : Ya[(2 * LSEQ - 1 - l1) * DDIM + d];
            dzs[t][d] = z1 - z0;
        }
    } else {
#pragma unroll
        for (int d = 0; d < DDIM; ++d) dzs[t][d] = 0.0f;  // pad row 511
    }
    krow[t] = 1.0f;   // K[0][:] = 1
    __syncthreads();

    // WMMA operand striping (ISA 7.12.2, 32-bit 16x4 A):
    //   lanes 0-15 : VGPR0=K0, VGPR1=K1 for M=lane ; lanes 16-31 : K2,K3 for M=lane-16
    // B (4x16) mirrors this with N in place of M; Gram B == A layout since B = dz^T.
    const int m16 = lane & 15;
    const int ksel = (int)(lane >> 4);        // 0 -> K{0,1}, 1 -> K{2,3}

    for (int i = 0; i < LL; ++i) {
        const int r = i & 15;
        if (r == 0) {
            // ---- compute Gram block inc[i..i+15][0..511] with fp32 WMMA ----
            const int rb = i;
            v2f a0, a1;
            a0.x = dzs[rb + m16][2 * ksel + 0];
            a0.y = dzs[rb + m16][2 * ksel + 1];
            a1.x = dzs[rb + m16][4 + 2 * ksel + 0];
            a1.y = dzs[rb + m16][4 + 2 * ksel + 1];
#pragma unroll
            for (int ct = 0; ct < 2; ++ct) {           // wave w -> column tiles 2w, 2w+1
                const int cb = ((int)wave * 2 + ct) * 16;
                v2f b0, b1;
                b0.x = dzs[cb + m16][2 * ksel + 0];
                b0.y = dzs[cb + m16][2 * ksel + 1];
                b1.x = dzs[cb + m16][4 + 2 * ksel + 0];
                b1.y = dzs[cb + m16][4 + 2 * ksel + 1];
                v8f c = {0.f, 0.f, 0.f, 0.f, 0.f, 0.f, 0.f, 0.f};
                // D = A(16x4) * B(4x16) + C, accumulated over the two K-halves of D=8
                c = __builtin_amdgcn_wmma_f32_16x16x4_f32(false, a0, false, b0,
                                                          (short)0, c, false, false);
                c = __builtin_amdgcn_wmma_f32_16x16x4_f32(false, a1, false, b1,
                                                          (short)0, c, false, false);
                // C/D layout: VGPR v holds (M = v + 8*(lane>=16), N = lane%16)
#pragma unroll
                for (int v = 0; v < 8; ++v) {
                    incBlk[v + 8 * ksel][cb + m16] = c[v];
                }
            }
            __syncthreads();
        }

        // ---- PDE row step as an inclusive scan ----
        // c_j = K[i][j+1] + K[i][j]*(inc[i][j]-1);  K[i+1][j+1] = 1 + prefix(c)_j
        float cval = 0.0f;
        if (t < LL) {
            const float e = incBlk[r][t];
            cval = krow[t + 1] + krow[t] * (e - 1.0f);
        }
        float s = cval;
#pragma unroll
        for (int off = 1; off < 32; off <<= 1) {
            float tmp = __shfl_up(s, (unsigned)off, 32);
            if (lane >= (unsigned)off) s += tmp;
        }
        if (lane == 31) waveSums[wave] = s;
        __syncthreads();                                   // also fences krow reads
        float woff = 0.0f;
        for (unsigned w = 0; w < wave; ++w) woff += waveSums[w];
        s += woff;
        if (t < LL) krow[t + 1] = 1.0f + s;                // krow[0] stays 1
        __syncthreads();
    }

    if (t == 0) {
        const float k = krow[LL];                          // K[511][511] = k(Z,Z)
        partial[a] = sqrtf(fmaxf(k - 1.0f, 0.0f));
    }
}

// Deterministic 64-way mean (no float atomics -> bit-stable across graph replays).
__global__ void SigLoss_reduce_kernel(const float* __restrict__ partial,
                                      float* __restrict__ out) {
    if (threadIdx.x == 0 && blockIdx.x == 0) {
        float s = 0.0f;
        for (int a = 0; a < A_BATCH; ++a) s += partial[a];
        out[0] = s * (1.0f / (float)A_BATCH);
    }
}

extern "C" void kernel_launch(void* const* d_in, const int* in_sizes, int n_in,
                              void* d_out, int out_size, void* d_ws, size_t ws_size,
                              hipStream_t stream) {
    const float* X = (const float*)d_in[0];   // (64, 256, 8) fp32
    const float* Y = (const float*)d_in[1];   // (64, 256, 8) fp32
    float* out = (float*)d_out;               // scalar fp32
    float* partial = (float*)d_ws;            // 64 floats of scratch

    SigLoss_pde_kernel<<<A_BATCH, NT, 0, stream>>>(X, Y, partial);
    SigLoss_reduce_kernel<<<1, 32, 0, stream>>>(partial, out);
}